// DynamicGraphConv_7121055777268
// MI455X (gfx1250) — compile-verified
//
#include <hip/hip_runtime.h>
#include <hip/hip_bf16.h>

// ---------------------------------------------------------------------------
// DynamicGraphConv (DGCNN EdgeConv) for MI455X / gfx1250, wave32 + WMMA f32.
//   B=8, C=64, N=4096, K=20, O=64
// Pipeline:
//   K0: transpose x (B,C,N)->xt (B,N,C) + row norms xx
//   K1: fused gram (v_wmma_f32_16x16x4_f32, double-buffered B, dual acc,
//       4 waves/block column-split, top-20 lists in LDS) -> idx
//   K2: pq = xt @ [w1a-w1b ; w1b]^T  (WMMA)   => h[bnk] = p[n] + q[idx]
//   K3: deterministic partial sums/sumsq per channel (BN stats)
//   K4: finalize BN scale/shift (s,t)
//   K5: fused normalize+ReLU + (160x64)x(64x64) WMMA GEMM + max over K
// ---------------------------------------------------------------------------

typedef __attribute__((ext_vector_type(2))) float v2f;
typedef __attribute__((ext_vector_type(8))) float v8f;

#define BB 8
#define CC 64
#define NN 4096
#define KK 20
#define OO 64
#define BN_EPS 1e-5f
#define NEG_BIG -3.402823466e38f

#define STAT_BLOCKS 640   // 655360 items / 1024 per block

__device__ __forceinline__ v8f wmma_f32_4(v2f a, v2f b, v8f c) {
  // D = A(16x4,f32) * B(4x16,f32) + C(16x16,f32)
  return __builtin_amdgcn_wmma_f32_16x16x4_f32(false, a, false, b, (short)0, c,
                                               false, false);
}

__device__ __forceinline__ void load_cols(const float* __restrict__ xcol,
                                          int coff, v2f (&bv)[16]) {
#pragma unroll
  for (int kb = 0; kb < 16; ++kb) {
    int c = kb * 4 + coff;
    v2f v; v.x = xcol[c]; v.y = xcol[c + 1];
    bv[kb] = v;
  }
}

// ---------------------------------------------------------------- K0
__global__ void k0_transpose_norm(const float* __restrict__ x,
                                  float* __restrict__ xt,
                                  float* __restrict__ xx) {
  int t = blockIdx.x * 256 + threadIdx.x;      // 0 .. B*N-1
  int b = t >> 12;
  int n = t & (NN - 1);
  const float* xp = x + (size_t)b * CC * NN + n;
  float* xo = xt + (size_t)t * CC;
  float acc = 0.f;
#pragma unroll
  for (int c = 0; c < CC; ++c) {
    float v = xp[(size_t)c * NN];
    xo[c] = v;
    acc += v * v;
  }
  xx[t] = acc;
}

// ---------------------------------------------------------------- K1
// 4 waves per block; each wave owns 64 of the 256 column tiles of one
// 16-row tile. Gram via WMMA f32 16x16x4 (double-buffered B, dual acc).
// Per-row top-20 lists live in LDS (register fast path: vmin/minpos only).
__global__ void __launch_bounds__(128, 1)
k1_knn(const float* __restrict__ xt,
       const float* __restrict__ xx,
       int* __restrict__ idxb) {
  int blk = blockIdx.x;                 // B * N/16 = 2048
  int b   = blk >> 8;                   // /256
  int n0  = (blk & 255) * 16;
  int tid  = threadIdx.x;               // 128 = 4 waves
  int wave = tid >> 5;
  int lane = tid & 31;
  int ln = lane & 15;
  bool hi = lane >= 16;
  int hi8 = hi ? 8 : 0;
  int coff = hi ? 2 : 0;

  __shared__ float dtile[4][16][17];    // 2*dot - xx_col
  __shared__ float tlv[4][16][KK];      // per-wave per-row top-20 values
  __shared__ int   tli[4][16][KK];      // ... and indices

  // init LDS top lists
  if (lane < 16) {
#pragma unroll
    for (int t = 0; t < KK; ++t) { tlv[wave][lane][t] = NEG_BIG; tli[wave][lane][t] = 0; }
  }

  // A operand: rows n0..n0+15, held in VGPRs for the whole sweep
  const float* xrow = xt + ((size_t)(b * NN + n0 + ln)) * CC;
  v2f a[16];
  load_cols(xrow, coff, a);

  float xxn = (lane < 16) ? xx[b * NN + n0 + lane] : 0.f;
  const float* xbase = xt + (size_t)b * NN * CC;
  const float* xxb   = xx + b * NN;

  float vmin = NEG_BIG;
  int   minpos = 0;

  int ct0 = wave * 64;                  // this wave's 64 column tiles
  v2f bv0[16], bv1[16];
  float xc0, xc1;
  load_cols(xbase + ((size_t)(ct0 * 16 + ln)) * CC, coff, bv0);
  xc0 = xxb[ct0 * 16 + ln];

#define K1_PROCESS(CT, BV, XC)                                                \
  {                                                                           \
    int m0 = (CT) * 16;                                                       \
    v8f acc0 = {0.f, 0.f, 0.f, 0.f, 0.f, 0.f, 0.f, 0.f};                      \
    v8f acc1 = {0.f, 0.f, 0.f, 0.f, 0.f, 0.f, 0.f, 0.f};                      \
    _Pragma("unroll")                                                         \
    for (int kb = 0; kb < 8; ++kb) {  /* two independent WMMA chains */       \
      acc0 = wmma_f32_4(a[2 * kb],     BV[2 * kb],     acc0);                 \
      acc1 = wmma_f32_4(a[2 * kb + 1], BV[2 * kb + 1], acc1);                 \
    }                                                                         \
    __builtin_amdgcn_wave_barrier();                                          \
    _Pragma("unroll")                                                         \
    for (int r = 0; r < 8; ++r)                                               \
      dtile[wave][r + hi8][ln] = 2.f * (acc0[r] + acc1[r]) - (XC);            \
    __builtin_amdgcn_wave_barrier();  /* DS ops of one wave run in order */   \
    if (lane < 16) {                                                          \
      _Pragma("unroll 4")                                                     \
      for (int j = 0; j < 16; ++j) {                                          \
        float d = dtile[wave][lane][j] - xxn;                                 \
        if (d > vmin) {                                                       \
          tlv[wave][lane][minpos] = d;                                        \
          tli[wave][lane][minpos] = m0 + j;                                   \
          float nv = tlv[wave][lane][0]; int np = 0;                          \
          _Pragma("unroll")                                                   \
          for (int t = 1; t < KK; ++t) {                                      \
            float tv = tlv[wave][lane][t];                                    \
            if (tv < nv) { nv = tv; np = t; }                                 \
          }                                                                   \
          vmin = nv; minpos = np;                                             \
        }                                                                     \
      }                                                                       \
    }                                                                         \
    __builtin_amdgcn_wave_barrier();                                          \
  }

  for (int i = 0; i < 64; i += 2) {
    // prefetch next tile while current tile computes/scans
    load_cols(xbase + ((size_t)((ct0 + i + 1) * 16 + ln)) * CC, coff, bv1);
    xc1 = xxb[(ct0 + i + 1) * 16 + ln];
    K1_PROCESS(ct0 + i, bv0, xc0);
    // overshoot at i==62 reads the next region of d_ws: harmless, unused
    load_cols(xbase + ((size_t)((ct0 + i + 2) * 16 + ln)) * CC, coff, bv0);
    xc0 = xxb[(ct0 + i + 2) * 16 + ln];
    K1_PROCESS(ct0 + i + 1, bv1, xc1);
  }
#undef K1_PROCESS

  __syncthreads();
  // wave 0 merges waves 1..3 lists into its own LDS list (exact top-20)
  if (wave == 0 && lane < 16) {
    float fmin = tlv[0][lane][0]; int fpos = 0;
#pragma unroll
    for (int t = 1; t < KK; ++t) {
      float tv = tlv[0][lane][t];
      if (tv < fmin) { fmin = tv; fpos = t; }
    }
    for (int w = 1; w < 4; ++w) {
#pragma unroll
      for (int t = 0; t < KK; ++t) {
        float d = tlv[w][lane][t];
        if (d > fmin) {
          tlv[0][lane][fpos] = d;
          tli[0][lane][fpos] = tli[w][lane][t];
          float nv = tlv[0][lane][0]; int np = 0;
#pragma unroll
          for (int u = 1; u < KK; ++u) {
            float tv = tlv[0][lane][u];
            if (tv < nv) { nv = tv; np = u; }
          }
          fmin = nv; fpos = np;
        }
      }
    }
    int* op = idxb + ((size_t)(b * NN + n0 + lane)) * KK;
#pragma unroll
    for (int t = 0; t < KK; ++t) op[t] = tli[0][lane][t];
  }
}

// ---------------------------------------------------------------- K2
// pq[m][0:64]=p=xt@(w1a-w1b)^T ; pq[m][64:128]=q=xt@w1b^T  (WMMA GEMM)
__global__ void __launch_bounds__(32, 1)
k2_pq(const float* __restrict__ xt,
      const float* __restrict__ w1,
      float* __restrict__ pq) {
  int m0 = blockIdx.x * 16;             // rows over B*N = 32768
  int lane = threadIdx.x;
  int ln = lane & 15;
  bool hi = lane >= 16;
  int hi8 = hi ? 8 : 0;
  int coff = hi ? 2 : 0;

  const float* xrow = xt + ((size_t)(m0 + ln)) * CC;
  v2f a[16];
  load_cols(xrow, coff, a);

  for (int jt = 0; jt < 8; ++jt) {
    int jj = jt * 16 + ln;              // output column 0..127
    v2f bv[16];
#pragma unroll
    for (int kb = 0; kb < 16; ++kb) {   // batch all B loads before the chain
      int c = kb * 4 + coff;
      float b0, b1;
      if (jj < 64) {                    // W = w1a - w1b
        b0 = w1[jj * 128 + c]     - w1[jj * 128 + 64 + c];
        b1 = w1[jj * 128 + c + 1] - w1[jj * 128 + 64 + c + 1];
      } else {                          // W = w1b
        b0 = w1[(jj - 64) * 128 + 64 + c];
        b1 = w1[(jj - 64) * 128 + 64 + c + 1];
      }
      v2f v; v.x = b0; v.y = b1;
      bv[kb] = v;
    }
    v8f acc0 = {0.f, 0.f, 0.f, 0.f, 0.f, 0.f, 0.f, 0.f};
    v8f acc1 = {0.f, 0.f, 0.f, 0.f, 0.f, 0.f, 0.f, 0.f};
#pragma unroll
    for (int kb = 0; kb < 8; ++kb) {
      acc0 = wmma_f32_4(a[2 * kb],     bv[2 * kb],     acc0);
      acc1 = wmma_f32_4(a[2 * kb + 1], bv[2 * kb + 1], acc1);
    }
#pragma unroll
    for (int r = 0; r < 8; ++r)
      pq[((size_t)(m0 + r + hi8)) * 128 + jt * 16 + ln] = acc0[r] + acc1[r];
  }
}

// ---------------------------------------------------------------- K3
// Partial per-channel sum & sumsq of h[bnk,c] = p[bn,c] + q[b,idx,c]
__global__ void k3_stats_partial(const float* __restrict__ pq,
                                 const int* __restrict__ idxb,
                                 float* __restrict__ partials) {
  int tid = threadIdx.x;          // 256
  int c = tid & 63;
  int slot = tid >> 6;            // 0..3
  int base = blockIdx.x * 1024;
  float s = 0.f, s2 = 0.f;
  for (int it = base + slot; it < base + 1024; it += 4) {
    int bn = it / KK;
    int b = bn >> 12;
    int j = idxb[it];
    float val = pq[(size_t)bn * 128 + c] +
                pq[((size_t)(b * NN + j)) * 128 + 64 + c];
    s += val;
    s2 += val * val;
  }
  __shared__ float red[256];
  red[tid] = s;
  __syncthreads();
  if (slot == 0) partials[blockIdx.x * 128 + c] =
      red[c] + red[64 + c] + red[128 + c] + red[192 + c];
  __syncthreads();
  red[tid] = s2;
  __syncthreads();
  if (slot == 0) partials[blockIdx.x * 128 + 64 + c] =
      red[c] + red[64 + c] + red[128 + c] + red[192 + c];
}

// ---------------------------------------------------------------- K4
// Finalize: s = gamma*rstd ; t = beta - mean*s
__global__ void k4_stats_final(const float* __restrict__ partials,
                               const float* __restrict__ gamma,
                               const float* __restrict__ beta,
                               float* __restrict__ snt) {
  int c = threadIdx.x;            // 64
  float S = 0.f, S2 = 0.f;
  for (int blk = 0; blk < STAT_BLOCKS; ++blk) {
    S  += partials[blk * 128 + c];
    S2 += partials[blk * 128 + 64 + c];
  }
  const float invM = 1.f / (float)((size_t)BB * NN * KK);
  float mean = S * invM;
  float var = S2 * invM - mean * mean;
  float rstd = rsqrtf(var + BN_EPS);
  float s = rstd * gamma[c];
  snt[c] = s;
  snt[64 + c] = beta[c] - mean * s;
}

// ---------------------------------------------------------------- K5
// Per block: 8 points (160 edge rows). normalize+ReLU -> LDS ->
// WMMA GEMM (160x64)x(64x64) -> LDS -> max over K -> d_out (B,O,N).
__global__ void __launch_bounds__(128, 1)
k5_out(const float* __restrict__ pq,
       const int* __restrict__ idxb,
       const float* __restrict__ snt,
       const float* __restrict__ w2,
       float* __restrict__ out) {
  int blk = blockIdx.x;                 // B * N/8 = 4096
  int b = blk >> 9;
  int nbase = (blk & 511) * 8;
  int tid = threadIdx.x;                // 128

  __shared__ float hn[160][68];         // padded: stride 68 -> no bank conflicts
  __shared__ float os[160][68];

  // Phase A: h_norm = relu((p+q)*s + t)
  {
    int c = tid & 63;
    float s = snt[c];
    float t = snt[64 + c];
    for (int r = tid >> 6; r < 160; r += 2) {
      int nloc = r / KK;
      int k = r - nloc * KK;
      int bn = b * NN + nbase + nloc;
      int j = idxb[(size_t)bn * KK + k];
      float val = pq[(size_t)bn * 128 + c] +
                  pq[((size_t)(b * NN + j)) * 128 + 64 + c];
      val = val * s + t;
      hn[r][c] = fmaxf(val, 0.f);
    }
  }
  __syncthreads();

  // Phase B: out_s = hn @ w2^T via WMMA (batched B, dual accumulators)
  {
    int wave = tid >> 5;
    int lane = tid & 31;
    int ln = lane & 15;
    bool hi = lane >= 16;
    int hi8 = hi ? 8 : 0;
    int coff = hi ? 2 : 0;
    for (int rt = wave; rt < 10; rt += 4) {
      int r0 = rt * 16;
      v2f a[16];
#pragma unroll
      for (int kb = 0; kb < 16; ++kb) {
        int c = kb * 4 + coff;
        v2f v; v.x = hn[r0 + ln][c]; v.y = hn[r0 + ln][c + 1];
        a[kb] = v;
      }
#pragma unroll
      for (int jt = 0; jt < 4; ++jt) {
        int o = jt * 16 + ln;
        v2f bv[16];
#pragma unroll
        for (int kb = 0; kb < 16; ++kb) {
          int c = kb * 4 + coff;
          v2f v; v.x = w2[o * 64 + c]; v.y = w2[o * 64 + c + 1];
          bv[kb] = v;
        }
        v8f acc0 = {0.f, 0.f, 0.f, 0.f, 0.f, 0.f, 0.f, 0.f};
        v8f acc1 = {0.f, 0.f, 0.f, 0.f, 0.f, 0.f, 0.f, 0.f};
#pragma unroll
        for (int kb = 0; kb < 8; ++kb) {
          acc0 = wmma_f32_4(a[2 * kb],     bv[2 * kb],     acc0);
          acc1 = wmma_f32_4(a[2 * kb + 1], bv[2 * kb + 1], acc1);
        }
#pragma unroll
        for (int r = 0; r < 8; ++r)
          os[r0 + r + hi8][jt * 16 + ln] = acc0[r] + acc1[r];
      }
    }
  }
  __syncthreads();

  // Phase C: max over K, store (B,O,N)
  {
    int c = tid & 63;
    for (int nl = tid >> 6; nl < 8; nl += 2) {
      float m = NEG_BIG;
#pragma unroll
      for (int k = 0; k < KK; ++k) m = fmaxf(m, os[nl * KK + k][c]);
      out[((size_t)(b * OO + c)) * NN + (nbase + nl)] = m;
    }
  }
}

// ---------------------------------------------------------------- launch
extern "C" void kernel_launch(void* const* d_in, const int* in_sizes, int n_in,
                              void* d_out, int out_size, void* d_ws, size_t ws_size,
                              hipStream_t stream) {
  const float* x     = (const float*)d_in[0];   // (8,64,4096)
  const float* w1    = (const float*)d_in[1];   // (64,128)
  const float* gamma = (const float*)d_in[2];   // (64)
  const float* beta  = (const float*)d_in[3];   // (64)
  const float* w2    = (const float*)d_in[4];   // (64,64)
  float* out = (float*)d_out;                   // (8,64,4096)

  // workspace carve-up (all offsets 16B-aligned)
  float* xt       = (float*)d_ws;                       // 2,097,152 f
  float* pq       = xt + (size_t)BB * NN * CC;          // 4,194,304 f
  float* xx       = pq + (size_t)BB * NN * 128;         //    32,768 f
  float* partials = xx + (size_t)BB * NN;               //    81,920 f
  float* snt      = partials + (size_t)STAT_BLOCKS * 128; //     128 f
  int*   idxb     = (int*)(snt + 128);                  //   655,360 i

  k0_transpose_norm<<<(BB * NN) / 256, 256, 0, stream>>>(x, xt, xx);
  k1_knn<<<BB * (NN / 16), 128, 0, stream>>>(xt, xx, idxb);
  k2_pq<<<(BB * NN) / 16, 32, 0, stream>>>(xt, w1, pq);
  k3_stats_partial<<<STAT_BLOCKS, 256, 0, stream>>>(pq, idxb, partials);
  k4_stats_final<<<1, 64, 0, stream>>>(partials, gamma, beta, snt);
  k5_out<<<BB * (NN / 8), 128, 0, stream>>>(pq, idxb, snt, w2, out);
}